// MySparseLayerSparseTensor_20555713479330
// MI455X (gfx1250) — compile-verified
//
#include <hip/hip_runtime.h>

// Block-sparse scatter GEMM for MI455X (gfx1250, wave32).
//   out[b, 4i+a] += sum_k W_e[a,k] * x[b, 4j+k]   for each edge e=(i,j)
//   out initialized to broadcast bias.
// Core op: V_WMMA_F32_16X16X4_F32 — K=4 matches the 4x4 block contraction.

typedef float v2f __attribute__((ext_vector_type(2)));
typedef float v8f __attribute__((ext_vector_type(8)));

// ---------------------------------------------------------------------------
// Kernel 1: out[b, s] = biases[s]  (vectorized float4 broadcast)
// ---------------------------------------------------------------------------
__global__ __launch_bounds__(256) void bias_init_kernel(
    float4* __restrict__ out4, const float4* __restrict__ bias4, int size4) {
  int col = blockIdx.x * blockDim.x + threadIdx.x;
  if (col < size4) {
    out4[(size_t)blockIdx.y * (size_t)size4 + col] = bias4[col];
  }
}

// ---------------------------------------------------------------------------
// Kernel 2: per-edge 4x4 block scatter via WMMA f32 16x16x4.
//
// One wave = one (edge, 64-batch group), 4 tiles of 16 batch rows:
//   A (16x4, f32): lane L (m = L&15, kh = L>>4) holds
//       VGPR0 = x[b0+m, 4j + 2*kh], VGPR1 = x[b0+m, 4j + 2*kh + 1]
//   B (4x16, f32): lane L (n = L&15) holds W[n, 2*kh + {0,1}]  (zero n>=4)
//   D (16x16): VGPR v -> (M=v, N=lane) / lanes 16-31 -> (M=v+8).
// Pipeline: 4 A-loads (clause) -> 4 WMMAs -> one masked 32-atomic scatter.
// KSIZE > 0 specializes the row stride at compile time so tile offsets
// become 24-bit instruction-immediate offsets (no v_mul_u64 per tile).
// ---------------------------------------------------------------------------
template <int KSIZE>
__global__ __launch_bounds__(256) void spmm_scatter_wmma_kernel(
    const float* __restrict__ x,
    const float* __restrict__ values,
    const int* __restrict__ indices,
    float* __restrict__ out,
    int sizeRt, unsigned bgShift, unsigned totalWork, unsigned totalWaves) {
  const int sz = (KSIZE > 0) ? KSIZE : sizeRt;
  const unsigned lane = threadIdx.x & 31u;
  const unsigned m = lane & 15u;    // A: batch row within tile; B/D: column N
  const unsigned kh = lane >> 4;    // K half (0 -> K{0,1}, 1 -> K{2,3})

  unsigned w = blockIdx.x * (blockDim.x >> 5) + (threadIdx.x >> 5);

  for (; w < totalWork; w += totalWaves) {
    const unsigned e = w >> bgShift;                  // edge id
    const unsigned bg = w & ((1u << bgShift) - 1u);   // 64-row batch group
    const unsigned base = e * 16u;                    // first nnz of edge block

    // First entry of the 4x4 block: (4*row[e], 4*col[e]) — wave-uniform.
    const int rowBase = indices[(size_t)base * 2u];
    const int colBase = indices[(size_t)base * 2u + 1u];

    // B = W^T padded with zero columns for n >= 4.
    v2f Bm = {0.0f, 0.0f};
    if (m < 4u) {
      Bm = *(const v2f*)(values + base + m * 4u + kh * 2u);
    }

    // Prefetch next work item's weights + indices (global_prefetch_b8).
    {
      const unsigned wn = w + totalWaves;
      if (wn < totalWork) {
        const unsigned bn = (wn >> bgShift) * 16u;
        __builtin_prefetch(values + bn, 0, 3);
        __builtin_prefetch(indices + (size_t)bn * 2u, 0, 3);
      }
    }

    const unsigned b0 = bg << 6;  // first of 64 batch rows for this wave
    const float* aPtr =
        x + (size_t)(b0 + m) * (size_t)sz + (size_t)colBase + (size_t)(kh * 2u);

    // Stage 1: all 4 A tiles (one b64 load each; constant offsets when KSIZE>0).
    v2f A[4];
#pragma unroll
    for (int t = 0; t < 4; ++t) {
      A[t] = *(const v2f*)(aPtr + (size_t)(t * 16) * (size_t)sz);
    }

    // Stage 2: 4 independent WMMAs (distinct D registers, shared B).
    v8f D[4];
#pragma unroll
    for (int t = 0; t < 4; ++t) {
      v8f C = {0.0f, 0.0f, 0.0f, 0.0f, 0.0f, 0.0f, 0.0f, 0.0f};
      // 8 args: (neg_a, A, neg_b, B, c_mod, C, reuse_a, reuse_b)
      D[t] = __builtin_amdgcn_wmma_f32_16x16x4_f32(
          false, A[t], false, Bm, (short)0, C, false, false);
    }

    // Stage 3: one masked scatter region. Valid lanes: 0-3 (M=v), 16-19 (M=v+8).
    if (m < 4u) {
      float* oPtr = out + (size_t)(b0 + 8u * kh) * (size_t)sz +
                    (size_t)(rowBase + (int)m);
#pragma unroll
      for (int t = 0; t < 4; ++t) {
        float* oT = oPtr + (size_t)(16 * t) * (size_t)sz;  // per-tile base
#pragma unroll
        for (int v = 0; v < 8; ++v) {
          atomicAdd(oT + (size_t)v * (size_t)sz, D[t][v]);  // v*sz*4B immediate
        }
      }
    }
  }
}

// ---------------------------------------------------------------------------
// Launch: inputs are {x[1024,40000] f32, values[5.12M] f32,
//                     biases[40000] f32, indices[5.12M,2] i32}
// ---------------------------------------------------------------------------
extern "C" void kernel_launch(void* const* d_in, const int* in_sizes, int n_in,
                              void* d_out, int out_size, void* d_ws,
                              size_t ws_size, hipStream_t stream) {
  const float* x = (const float*)d_in[0];
  const float* values = (const float*)d_in[1];
  const float* biases = (const float*)d_in[2];
  const int* indices = (const int*)d_in[3];
  float* out = (float*)d_out;

  const int size = in_sizes[2];            // 40000
  const int batch = in_sizes[0] / size;    // 1024 (multiple of 64 assumed)
  const int nEdges = in_sizes[1] / 16;     // 320000

  // out = bias broadcast (size % 4 == 0).
  {
    const int size4 = size / 4;
    dim3 block(256);
    dim3 grid((unsigned)((size4 + 255) / 256), (unsigned)batch);
    bias_init_kernel<<<grid, block, 0, stream>>>((float4*)out,
                                                 (const float4*)biases, size4);
  }

  // Scatter kernel: persistent waves over (edge x 64-batch-group) work items.
  int nBG = batch >> 6;                    // groups of 64 batch rows
  if (nBG < 1) nBG = 1;
  const unsigned bgShift = (unsigned)(31 - __builtin_clz((unsigned)nBG));
  const unsigned totalWork = (unsigned)nEdges << bgShift;   // 5.12M work items
  const unsigned wavesPerBlock = 8;                         // 256 threads, wave32
  unsigned blocks = 4096;                                   // 32768 persistent waves
  unsigned totalWaves = blocks * wavesPerBlock;
  if (totalWaves > totalWork) {
    blocks = (totalWork + wavesPerBlock - 1) / wavesPerBlock;
    totalWaves = blocks * wavesPerBlock;
  }

  if (size == 40000) {
    spmm_scatter_wmma_kernel<40000><<<dim3(blocks), dim3(256), 0, stream>>>(
        x, values, indices, out, size, bgShift, totalWork, totalWaves);
  } else {
    spmm_scatter_wmma_kernel<0><<<dim3(blocks), dim3(256), 0, stream>>>(
        x, values, indices, out, size, bgShift, totalWork, totalWaves);
  }
}